// DotProductAttention_33784212750985
// MI455X (gfx1250) — compile-verified
//
#include <hip/hip_runtime.h>
#include <stdint.h>

// CDNA5 / gfx1250: wave32, WMMA 16x16 shapes. We use the f32 WMMA
// (V_WMMA_F32_16X16X4_F32) since the workload is HBM-bound on the 536MB
// attn-matrix store, so f32 matrix math is free and numerics-exact.

typedef __attribute__((ext_vector_type(2))) float v2f;
typedef __attribute__((ext_vector_type(8))) float v8f;

#if defined(__HIP_DEVICE_COMPILE__)
#if !__has_builtin(__builtin_amdgcn_wmma_f32_16x16x4_f32)
#error "gfx1250 f32 WMMA builtin __builtin_amdgcn_wmma_f32_16x16x4_f32 not available"
#endif
#endif

#define B_ 8
#define S_ 4096
#define D_ 64

static constexpr int WAVES_PER_BLOCK = 4;                 // 128 threads
static constexpr int ROWS_PER_BLOCK  = WAVES_PER_BLOCK * 16; // 64 q-rows/block
static constexpr int P_STRIDE        = 18;                // pad: keeps b64 LDS loads 8B-aligned

__global__ __launch_bounds__(WAVES_PER_BLOCK * 32)
void DotProductAttention_33784212750985_kernel(const float* __restrict__ Q,
                                               const float* __restrict__ K,
                                               const float* __restrict__ V,
                                               const unsigned char* __restrict__ mask,
                                               float* __restrict__ ctx_out,
                                               float* __restrict__ attn_out)
{
    // Per-wave 16x16 P-tile staging (C-layout -> A-layout reshuffle via LDS).
    __shared__ float p_lds[WAVES_PER_BLOCK][16 * P_STRIDE];

    const int tid  = threadIdx.x;
    const int wave = tid >> 5;
    const int lane = tid & 31;
    const int lrow = lane & 15;   // row index for A/B operand loads; col index (N) in C-layout
    const int hi   = lane >> 4;   // half-wave select
    const int ks   = hi * 2;      // K-offset pair for A/B operands (f32 16x16x4 layout)

    const int blocks_per_batch = S_ / ROWS_PER_BLOCK;     // 64
    const int b  = blockIdx.x / blocks_per_batch;
    const int q0 = (blockIdx.x % blocks_per_batch) * ROWS_PER_BLOCK + wave * 16;

    const float* __restrict__ Qb = Q + (size_t)b * S_ * D_;
    const float* __restrict__ Kb = K + (size_t)b * S_ * D_;
    const float* __restrict__ Vb = V + (size_t)b * S_ * D_;
    const unsigned char* __restrict__ Mb = mask + (size_t)b * S_ * S_;

    // ---- Preload the Q strip in WMMA A-layout: lane holds row q0+lrow,
    //      elements {4j+ks, 4j+ks+1} for j=0..15 (32 VGPRs, reused both sweeps).
    v2f qa[16];
    {
        const float* qrow = Qb + (size_t)(q0 + lrow) * D_;
        #pragma unroll
        for (int j = 0; j < 16; ++j)
            qa[j] = *(const v2f*)(qrow + 4 * j + ks);
    }

    // ================= Sweep 1: per-lane online softmax stats =================
    // Each lane owns column (k0+lrow) of rows {r+8*hi}; keep per-lane running
    // (max, rescaled-sum) over its own column stream -> no shuffles in the loop.
    float m_run[8], l_run[8];
    #pragma unroll
    for (int r = 0; r < 8; ++r) { m_run[r] = -3.0e38f; l_run[r] = 0.0f; }

    for (int k0 = 0; k0 < S_; k0 += 16) {
        const float* krow = Kb + (size_t)(k0 + lrow) * D_;
        v8f acc = {};
        #pragma unroll
        for (int j = 0; j < 16; ++j) {
            v2f kb = *(const v2f*)(krow + 4 * j + ks);
            acc = __builtin_amdgcn_wmma_f32_16x16x4_f32(
                false, qa[j], false, kb, (short)0, acc, false, false);
        }
        #pragma unroll
        for (int r = 0; r < 8; ++r) {
            const int qr = r + 8 * hi;                       // C-layout row
            float s = acc[r] * 0.125f;                       // 1/sqrt(64)
            if (Mb[(size_t)(q0 + qr) * S_ + k0 + lrow]) s = -10000.0f;
            float mo = m_run[r];
            float mn = fmaxf(mo, s);
            l_run[r] = l_run[r] * __expf(mo - mn) + __expf(s - mn);
            m_run[r] = mn;
        }
    }

    // Cross-lane merge within each 16-lane half (xor 1,2,4,8 stays in-half).
    #pragma unroll
    for (int r = 0; r < 8; ++r) {
        float m = m_run[r], l = l_run[r];
        #pragma unroll
        for (int xm = 1; xm <= 8; xm <<= 1) {
            float mo = __shfl_xor(m, xm, 32);
            float lo = __shfl_xor(l, xm, 32);
            float mn = fmaxf(m, mo);
            l = l * __expf(m - mn) + lo * __expf(mo - mn);
            m = mn;
        }
        m_run[r] = m;
        l_run[r] = 1.0f / l;   // store reciprocal of the softmax denominator
    }

    // ================= Sweep 2: emit attn, accumulate context =================
    v8f cacc[4];
    #pragma unroll
    for (int nc = 0; nc < 4; ++nc) cacc[nc] = (v8f){};

    float* my_p = p_lds[wave];

    for (int k0 = 0; k0 < S_; k0 += 16) {
        // Recompute score tile (K streamed from L2; recompute beats a 1 GB
        // round-trip of unnormalized exp through HBM).
        const float* krow = Kb + (size_t)(k0 + lrow) * D_;
        v8f acc = {};
        #pragma unroll
        for (int j = 0; j < 16; ++j) {
            v2f kb = *(const v2f*)(krow + 4 * j + ks);
            acc = __builtin_amdgcn_wmma_f32_16x16x4_f32(
                false, qa[j], false, kb, (short)0, acc, false, false);
        }

        __syncthreads();   // fence previous iteration's LDS reads vs new writes
        #pragma unroll
        for (int r = 0; r < 8; ++r) {
            const int qr = r + 8 * hi;
            float s = acc[r] * 0.125f;
            if (Mb[(size_t)(q0 + qr) * S_ + k0 + lrow]) s = -10000.0f;
            float p = __expf(s - m_run[r]) * l_run[r];
            attn_out[((size_t)b * S_ + q0 + qr) * S_ + k0 + lrow] = p;
            my_p[qr * P_STRIDE + lrow] = p;                  // stage C-layout tile
        }
        __syncthreads();   // P tile visible before A-layout reload

        // ctx(16x64) += P(16x16) * V(16x64), as 4 N-chunks x 4 K-chunks of WMMA.
        #pragma unroll
        for (int j = 0; j < 4; ++j) {
            // A-layout reload of P: lane = row lrow, cols {4j+ks, 4j+ks+1}
            v2f pa = *(const v2f*)(&my_p[lrow * P_STRIDE + 4 * j + ks]);
            #pragma unroll
            for (int nc = 0; nc < 4; ++nc) {
                const float* vcol = Vb + (size_t)(k0 + 4 * j + ks) * D_ + nc * 16 + lrow;
                v2f vb2;
                vb2.x = vcol[0];       // B[k][n] = V[k0+4j+ks  ][nc*16+lrow]
                vb2.y = vcol[D_];      // B[k+1][n]
                cacc[nc] = __builtin_amdgcn_wmma_f32_16x16x4_f32(
                    false, pa, false, vb2, (short)0, cacc[nc], false, false);
            }
        }
    }

    // ---- Write context tile (C-layout scatter, 16-float contiguous runs). ----
    #pragma unroll
    for (int nc = 0; nc < 4; ++nc) {
        #pragma unroll
        for (int r = 0; r < 8; ++r) {
            const int qr = r + 8 * hi;
            ctx_out[((size_t)b * S_ + q0 + qr) * D_ + nc * 16 + lrow] = cacc[nc][r];
        }
    }
}

extern "C" void kernel_launch(void* const* d_in, const int* in_sizes, int n_in,
                              void* d_out, int out_size, void* d_ws, size_t ws_size,
                              hipStream_t stream) {
    (void)in_sizes; (void)n_in; (void)out_size; (void)d_ws; (void)ws_size;
    const float* q = (const float*)d_in[0];
    const float* k = (const float*)d_in[1];
    const float* v = (const float*)d_in[2];
    const unsigned char* m = (const unsigned char*)d_in[3];   // jax bool -> 1 byte

    float* ctx  = (float*)d_out;                               // [B,S,D] first
    float* attn = ctx + (size_t)B_ * S_ * D_;                  // then [B,S,S]

    const int grid = B_ * (S_ / ROWS_PER_BLOCK);               // 512 blocks
    DotProductAttention_33784212750985_kernel<<<grid, WAVES_PER_BLOCK * 32, 0, stream>>>(
        q, k, v, m, ctx, attn);
}